// CapsuleLayer_74809740362085
// MI455X (gfx1250) — compile-verified
//
#include <hip/hip_runtime.h>
#include <cstdint>

typedef __attribute__((ext_vector_type(2))) float v2f;
typedef __attribute__((ext_vector_type(8))) float v8f;

#define B_TOT 64
#define N_IN  2048
#define D_IN  16
#define N_OUT 32
#define D_OUT 32
#define BT    16   // b rows per workgroup (one WMMA M tile)
#define CHUNK 16   // n values per workgroup

// dynamic LDS layout (in floats)
#define LDS_BLOG 0                 // [16][32]
#define LDS_C    512               // [16][32]
#define LDS_SMAX 1024              // [16]
#define LDS_SINV 1040              // [16]
#define LDS_KBUF 1056              // [2 bufs][8 waves][4 j][16 k][32 d]
#define KBUF_WAVE_FLOATS 2048      // 4*16*32
#define KBUF_BUF_FLOATS  16384     // 8 waves * 2048
#define LDS_TOTAL_FLOATS (LDS_KBUF + 2 * KBUF_BUF_FLOATS)  // 33824 -> 135296 B

__global__ void zero_f32(float* __restrict__ p, int n) {
  int i = blockIdx.x * blockDim.x + threadIdx.x;
  int stride = gridDim.x * blockDim.x;
  for (; i < n; i += stride) p[i] = 0.f;
}

// xT[n][b][k] = x[b][n][k]  (per-n x tiles contiguous: 16b x 16k = 1KB per btile)
__global__ void transpose_x(const float* __restrict__ x, float* __restrict__ xT) {
  int o = blockIdx.x * blockDim.x + threadIdx.x;
  if (o >= B_TOT * N_IN * D_IN) return;
  int k = o & 15;
  int b = (o >> 4) & 63;
  int n = o >> 10;
  xT[o] = x[((size_t)b * N_IN + n) * D_IN + k];
}

// issue async copy of one 8KB kernel[n] slice (this wave's 4 j's) into LDS
__device__ __forceinline__ void stage_tile_async(const float* __restrict__ gsrc,
                                                 unsigned lds_base_bytes, int lane) {
#pragma unroll
  for (int c = 0; c < 16; ++c) {
    unsigned loff = lds_base_bytes + (unsigned)((c * 32 + lane) * 16);
    const float* g = gsrc + (size_t)(c * 32 + lane) * 4;
    asm volatile("global_load_async_to_lds_b128 %0, %1, off"
                 :: "v"(loff), "v"((unsigned long long)(uintptr_t)g)
                 : "memory");
  }
}

// One routing iteration, fused: recompute U via f32 WMMA (B staged in LDS by the
// async engine, double buffered), agreement dot, Blog update, softmax over j,
// weighted S accumulation + atomic flush.
// mode 0: iter0 (C = 1/32 uniform, no Blog/V use)
// mode 1: update Blog (read+write), softmax, accumulate
// mode 2: like 1 but skip the (dead) global Blog writeback
__global__ __launch_bounds__(256) void caps_route(
    const float* __restrict__ xT,    // [N_IN][64][16]
    const float* __restrict__ kern,  // [N_IN][32][16][32]
    const float* __restrict__ V,     // [64][32][32] previous V (unused mode 0)
    float* __restrict__ Blog,        // [64][N_IN][32]
    float* __restrict__ S,           // [64][32][32] global accumulators
    int mode)
{
  extern __shared__ float smem[];
  float* blog_lds = smem + LDS_BLOG;
  float* c_lds    = smem + LDS_C;
  float* smax     = smem + LDS_SMAX;
  float* sinv     = smem + LDS_SINV;

  const int tid  = threadIdx.x;
  const int wave = tid >> 5;
  const int lane = tid & 31;
  const int hi   = lane >> 4;    // rows b = r + 8*hi, cols d = dt*16 + lm
  const int lm   = lane & 15;
  const int bt   = blockIdx.x & 3;          // which b-tile of 16
  const int n0   = (blockIdx.x >> 2) * CHUNK;
  const int j0   = wave * 4;                // each wave owns 4 output capsules

  // byte base of this wave's two staging buffers
  const unsigned kbuf0 =
      (unsigned)(uintptr_t)(smem + LDS_KBUF + wave * KBUF_WAVE_FLOATS);
  const unsigned kbuf_stride = KBUF_BUF_FLOATS * 4;  // bytes between buffers

  v8f sacc[4][2];
  {
    v8f zc = {};
    for (int jj = 0; jj < 4; ++jj)
      for (int dt = 0; dt < 2; ++dt) sacc[jj][dt] = zc;
  }

  // hoist V fragments: V[b,j,d] is n-invariant (loaded once per kernel)
  float vfrag[4][2][8];
  if (mode != 0) {
#pragma unroll
    for (int jj = 0; jj < 4; ++jj) {
      const float* vj = V + (size_t)(j0 + jj) * D_OUT;
#pragma unroll
      for (int r = 0; r < 8; ++r) {
        int b = bt * BT + r + 8 * hi;
        vfrag[jj][0][r] = vj[(size_t)b * N_OUT * D_OUT + lm];
        vfrag[jj][1][r] = vj[(size_t)b * N_OUT * D_OUT + 16 + lm];
      }
    }
  }

  // prologue: stage tile for n0
  stage_tile_async(kern + ((size_t)n0 * N_OUT + j0) * D_IN * D_OUT,
                   kbuf0 + (n0 & 1) * kbuf_stride, lane);

  for (int n = n0; n < n0 + CHUNK; ++n) {
    // stage next tile, then wait for current tile (asyncs complete in order)
    if (n + 1 < n0 + CHUNK) {
      stage_tile_async(kern + ((size_t)(n + 1) * N_OUT + j0) * D_IN * D_OUT,
                       kbuf0 + ((n + 1) & 1) * kbuf_stride, lane);
      asm volatile("s_wait_asynccnt 16" ::: "memory");
    } else {
      asm volatile("s_wait_asynccnt 0" ::: "memory");
    }

    // A fragments straight from global (xT tile is 1KB, L2-resident)
    const float* xtile = xT + ((size_t)n * B_TOT + bt * BT) * D_IN;
    v2f a[4];
#pragma unroll
    for (int ks = 0; ks < 4; ++ks) {
      int kk = 4 * ks + 2 * hi;
      a[ks].x = xtile[lm * D_IN + kk];
      a[ks].y = xtile[lm * D_IN + kk + 1];
    }

    // B fragments from wave-private LDS staging buffer: [jj][k][d]
    const float* kw = smem + LDS_KBUF + (n & 1) * KBUF_BUF_FLOATS +
                      wave * KBUF_WAVE_FLOATS;

    v8f u[4][2];
#pragma unroll
    for (int jj = 0; jj < 4; ++jj) {
      v2f bfr[2][4];
#pragma unroll
      for (int dt = 0; dt < 2; ++dt) {
#pragma unroll
        for (int ks = 0; ks < 4; ++ks) {
          int kk = 4 * ks + 2 * hi;
          bfr[dt][ks].x = kw[(jj * D_IN + kk) * D_OUT + dt * 16 + lm];
          bfr[dt][ks].y = kw[(jj * D_IN + kk + 1) * D_OUT + dt * 16 + lm];
        }
      }
#pragma unroll
      for (int dt = 0; dt < 2; ++dt) {
        v8f acc = {};
#pragma unroll
        for (int ks = 0; ks < 4; ++ks) {
          acc = __builtin_amdgcn_wmma_f32_16x16x4_f32(
              false, a[ks], false, bfr[dt][ks], (short)0, acc, false, false);
        }
        u[jj][dt] = acc;
      }
    }

    if (mode != 0) {
      // agreement t[b] = sum_d U[b,j,d]*V[b,j,d], then Blog update into LDS
#pragma unroll
      for (int jj = 0; jj < 4; ++jj) {
        int j = j0 + jj;
#pragma unroll
        for (int r = 0; r < 8; ++r) {
          float pv = u[jj][0][r] * vfrag[jj][0][r] + u[jj][1][r] * vfrag[jj][1][r];
          // reduce across the 16 lanes of each half (halves are independent b's)
          pv += __shfl_xor(pv, 1);
          pv += __shfl_xor(pv, 2);
          pv += __shfl_xor(pv, 4);
          pv += __shfl_xor(pv, 8);
          if (lm == r) {  // lanes r and 16+r commit rows r and r+8
            int b = bt * BT + r + 8 * hi;
            size_t bi = ((size_t)b * N_IN + n) * N_OUT + j;
            float nb = Blog[bi] + pv;
            if (mode == 1) Blog[bi] = nb;
            blog_lds[(r + 8 * hi) * N_OUT + j] = nb;
          }
        }
      }
      __syncthreads();
      // softmax stats over j per b
      if (tid < BT) {
        float m = -3.4e38f;
        for (int j = 0; j < N_OUT; ++j) m = fmaxf(m, blog_lds[tid * N_OUT + j]);
        float s = 0.f;
        for (int j = 0; j < N_OUT; ++j) s += __expf(blog_lds[tid * N_OUT + j] - m);
        smax[tid] = m;
        sinv[tid] = 1.f / s;
      }
      __syncthreads();
      for (int e = tid; e < BT * N_OUT; e += 256) {
        int b = e >> 5;
        c_lds[e] = __expf(blog_lds[e] - smax[b]) * sinv[b];
      }
      __syncthreads();
    }

    // S[b,j,d] += C[b,n,j] * U[b,n,j,d], accumulated in WMMA fragment layout
#pragma unroll
    for (int jj = 0; jj < 4; ++jj) {
      int j = j0 + jj;
#pragma unroll
      for (int r = 0; r < 8; ++r) {
        float c = (mode == 0) ? 0.03125f : c_lds[(r + 8 * hi) * N_OUT + j];
        sacc[jj][0][r] += c * u[jj][0][r];
        sacc[jj][1][r] += c * u[jj][1][r];
      }
    }
  }

  // flush per-WG S partials (one atomic per element per WG)
#pragma unroll
  for (int jj = 0; jj < 4; ++jj) {
    int j = j0 + jj;
#pragma unroll
    for (int dt = 0; dt < 2; ++dt) {
#pragma unroll
      for (int r = 0; r < 8; ++r) {
        int b = bt * BT + r + 8 * hi;
        unsafeAtomicAdd(&S[((size_t)b * N_OUT + j) * D_OUT + dt * 16 + lm],
                        sacc[jj][dt][r]);
      }
    }
  }
}

__global__ void squash_k(const float* __restrict__ S, float* __restrict__ V) {
  int i = blockIdx.x * blockDim.x + threadIdx.x;  // one (b,j) vector each
  if (i >= B_TOT * N_OUT) return;
  const float* s = S + (size_t)i * D_OUT;
  float sq = 0.f;
#pragma unroll
  for (int d = 0; d < D_OUT; ++d) sq += s[d] * s[d];
  float scale = (sq / (1.f + sq)) * rsqrtf(sq + 1e-8f);
  float* v = V + (size_t)i * D_OUT;
#pragma unroll
  for (int d = 0; d < D_OUT; ++d) v[d] = s[d] * scale;
}

extern "C" void kernel_launch(void* const* d_in, const int* in_sizes, int n_in,
                              void* d_out, int out_size, void* d_ws, size_t ws_size,
                              hipStream_t stream) {
  const float* x    = (const float*)d_in[0];
  const float* kern = (const float*)d_in[1];
  float* out = (float*)d_out;

  float* xT   = (float*)d_ws;                               // 8 MB
  float* Blog = xT + (size_t)N_IN * B_TOT * D_IN;           // 16 MB
  float* S    = Blog + (size_t)B_TOT * N_IN * N_OUT;        // 256 KB

  dim3 blk(256);
  dim3 grid(4 * (N_IN / CHUNK));  // 4 b-tiles x 128 n-chunks = 512 WGs
  int sqBlocks = (B_TOT * N_OUT + 255) / 256;
  size_t smemBytes = LDS_TOTAL_FLOATS * sizeof(float);  // ~132 KB dynamic LDS

  transpose_x<<<(B_TOT * N_IN * D_IN + 255) / 256, blk, 0, stream>>>(x, xT);
  zero_f32<<<1024, blk, 0, stream>>>(Blog, B_TOT * N_IN * N_OUT);
  zero_f32<<<64, blk, 0, stream>>>(S, B_TOT * N_OUT * D_OUT);

  // iteration 0: C uniform
  caps_route<<<grid, blk, smemBytes, stream>>>(xT, kern, out, Blog, S, 0);
  squash_k<<<sqBlocks, blk, 0, stream>>>(S, out);

  // iteration 1: Blog1 = 0 + U.V0 ; C = softmax(Blog1)
  zero_f32<<<64, blk, 0, stream>>>(S, B_TOT * N_OUT * D_OUT);
  caps_route<<<grid, blk, smemBytes, stream>>>(xT, kern, out, Blog, S, 1);
  squash_k<<<sqBlocks, blk, 0, stream>>>(S, out);

  // iteration 2: Blog2 = Blog1 + U.V1 (consumed in-kernel, no writeback)
  zero_f32<<<64, blk, 0, stream>>>(S, B_TOT * N_OUT * D_OUT);
  caps_route<<<grid, blk, smemBytes, stream>>>(xT, kern, out, Blog, S, 2);
  squash_k<<<sqBlocks, blk, 0, stream>>>(S, out);
}